// PINN_12154757448404
// MI455X (gfx1250) — compile-verified
//
#include <hip/hip_runtime.h>
#include <stdint.h>

#define BLOCK 256

typedef float v2f __attribute__((ext_vector_type(2)));
typedef float v8f __attribute__((ext_vector_type(8)));

__device__ __forceinline__ float fast_tanh(float x) {
    // tanh(x) = sign(x) * (1 - e) / (1 + e),  e = exp(-2|x|)  (overflow-safe)
    float ax = __builtin_fabsf(x);
    float e  = __expf(-2.0f * ax);
    float r  = __fdividef(1.0f - e, 1.0f + e);
    return __builtin_copysignf(r, x);
}

// SWAPX16: group-of-32 swizzle, xor_mask=0x10, and_mask=0x1f -> lane L <-> lane L^16
__device__ __forceinline__ float swap16(float v) {
    return __int_as_float(__builtin_amdgcn_ds_swizzle(__float_as_int(v), 0x401F));
}

__global__ __launch_bounds__(BLOCK) void pinn_pde_kernel(
    const float* __restrict__ X,
    const float* __restrict__ gW1, const float* __restrict__ gb1,
    const float* __restrict__ gW2, const float* __restrict__ gb2,
    const float* __restrict__ gW3, const float* __restrict__ gb3,
    const float* __restrict__ gW4, const float* __restrict__ gb4,
    const float* __restrict__ gW5,
    float* __restrict__ out, int V)
{
    __shared__ float sx[3 * BLOCK];
    const int       tid   = threadIdx.x;
    const long long pbase = (long long)blockIdx.x * BLOCK;
    const long long fbase = pbase * 3;
    const long long ftot  = (long long)V * 3;

    // ---- async-copy stage: 768 contiguous floats -> LDS (coalesced) ----
    long long i0 = fbase + tid;
    long long i1 = i0 + BLOCK;
    long long i2 = i1 + BLOCK;
    if (i0 >= ftot) i0 = ftot - 1;
    if (i1 >= ftot) i1 = ftot - 1;
    if (i2 >= ftot) i2 = ftot - 1;

    unsigned     l0 = (unsigned)(uintptr_t)(&sx[tid]);   // LDS byte offset (addr[31:0])
    unsigned     l1 = l0 + 4u * BLOCK;
    unsigned     l2 = l1 + 4u * BLOCK;
    const float* p0 = X + i0;
    const float* p1 = X + i1;
    const float* p2 = X + i2;

    asm volatile(
        "global_load_async_to_lds_b32 %0, %3, off\n\t"
        "global_load_async_to_lds_b32 %1, %4, off\n\t"
        "global_load_async_to_lds_b32 %2, %5, off\n\t"
        "s_wait_asynccnt 0"
        :
        : "v"(l0), "v"(l1), "v"(l2), "v"(p0), "v"(p1), "v"(p2)
        : "memory");

    // ---- W2 in WMMA A-matrix layout (M=l rows across lanes, K=m split per half) ----
    // A 16x4 f32 layout: lane L holds row M=L&15; VGPR0 = K=2*(L/16), VGPR1 = K=2*(L/16)+1.
    const int  lane  = tid & 31;
    const int  arow  = lane & 15;
    const int  khalf = (lane >> 4) & 1;
    const bool lo    = (lane < 16);
    v2f Afrag[3];
#pragma unroll
    for (int c = 0; c < 3; ++c) {
        int r  = arow > 9 ? 9 : arow;          // clamp OOB rows (results unread)
        int k0 = 4 * c + 2 * khalf;
        int k1 = k0 + 1;
        if (k0 > 9) k0 = 9;                    // cols 10,11 hit zeroed B rows
        if (k1 > 9) k1 = 9;
        Afrag[c].x = gW2[10 * r + k0];
        Afrag[c].y = gW2[10 * r + k1];
    }

    __syncthreads();

    const float xin0   = sx[3 * tid + 0];
    const float xin1   = sx[3 * tid + 1];
    const float tcoord = sx[3 * tid + 2];

    // ---- wave-uniform weights (s_load -> SGPRs, confirmed in round-1 asm) ----
    float W1[10][3], b1[10];
#pragma unroll
    for (int m = 0; m < 10; ++m) {
        b1[m] = gb1[m];
#pragma unroll
        for (int n = 0; n < 3; ++n) W1[m][n] = gW1[3 * m + n];
    }
    float W2[10][10], b2[10];
#pragma unroll
    for (int l = 0; l < 10; ++l) {
        b2[l] = gb2[l];
#pragma unroll
        for (int m = 0; m < 10; ++m) W2[l][m] = gW2[10 * l + m];
    }
    float W3[5][10], b3[5];
#pragma unroll
    for (int k = 0; k < 5; ++k) {
        b3[k] = gb3[k];
#pragma unroll
        for (int l = 0; l < 10; ++l) W3[k][l] = gW3[10 * k + l];
    }
    float W4[5][5], b4[5];
#pragma unroll
    for (int j = 0; j < 5; ++j) {
        b4[j] = gb4[j];
#pragma unroll
        for (int k = 0; k < 5; ++k) W4[j][k] = gW4[5 * j + k];
    }
    float W5[5];
#pragma unroll
    for (int j = 0; j < 5; ++j) W5[j] = gW5[j];

    // ---- layer 1: z1 = W1 x + b1 ; x1 = sin, S1 = cos ----
    float x1[10], S1[10];
#pragma unroll
    for (int m = 0; m < 10; ++m) {
        float z = b1[m] + W1[m][0] * xin0 + W1[m][1] * xin1 + W1[m][2] * tcoord;
        x1[m] = __sinf(z);
        S1[m] = __cosf(z);
    }
    // VMN[m][n] = S1[m] * W1[m][n]
    float VMN[10][3];
#pragma unroll
    for (int m = 0; m < 10; ++m)
#pragma unroll
        for (int n = 0; n < 3; ++n) VMN[m][n] = S1[m] * W1[m][n];

    // ================= A2 = W2 @ VMN via V_WMMA_F32_16X16X4_F32 =================
    // Two tiles per wave: N = points {lane 0..15} and {lane 16..31}; M = l; K = m (10->12).
    // B 4x16 layout: VGPR0 holds rows {K, K+2} split across lane halves, VGPR1 rows {K+1, K+3}.
    float A2[10][3];
#pragma unroll
    for (int n = 0; n < 3; ++n) {
        float sw[10];
#pragma unroll
        for (int m = 0; m < 10; ++m) sw[m] = swap16(VMN[m][n]);

        v2f B0[3], B1[3];
        B0[0].x = lo ? VMN[0][n] : sw[2];   B0[0].y = lo ? VMN[1][n] : sw[3];
        B1[0].x = lo ? sw[0] : VMN[2][n];   B1[0].y = lo ? sw[1] : VMN[3][n];
        B0[1].x = lo ? VMN[4][n] : sw[6];   B0[1].y = lo ? VMN[5][n] : sw[7];
        B1[1].x = lo ? sw[4] : VMN[6][n];   B1[1].y = lo ? sw[5] : VMN[7][n];
        B0[2].x = lo ? VMN[8][n] : 0.0f;    B0[2].y = lo ? VMN[9][n] : 0.0f;  // K=10,11 -> 0
        B1[2].x = lo ? sw[8] : 0.0f;        B1[2].y = lo ? sw[9] : 0.0f;

        v8f c0 = {};
        v8f c1 = {};
#pragma unroll
        for (int c = 0; c < 3; ++c) {
            c0 = __builtin_amdgcn_wmma_f32_16x16x4_f32(false, Afrag[c], false, B0[c],
                                                       (short)0, c0, false, false);
            c1 = __builtin_amdgcn_wmma_f32_16x16x4_f32(false, Afrag[c], false, B1[c],
                                                       (short)0, c1, false, false);
        }
        // C layout: vgpr r = rows {r, r+8} split across lane halves; gather to per-thread.
#pragma unroll
        for (int l = 0; l < 8; ++l) A2[l][n] = lo ? c0[l] : swap16(c1[l]);
#pragma unroll
        for (int l = 8; l < 10; ++l) A2[l][n] = lo ? swap16(c0[l - 8]) : c1[l - 8];
    }
    // ============================================================================

    // ---- layer 2 activations (per-thread) ----
    float x2[10], S2[10];
#pragma unroll
    for (int l = 0; l < 10; ++l) {
        float z = b2[l];
#pragma unroll
        for (int m = 0; m < 10; ++m) z += W2[l][m] * x1[m];
        float th = fast_tanh(z);
        x2[l] = th; S2[l] = 1.0f - th * th;
    }
    float B2[10][3];                       // B2 = S2 ⊙ A2  (scaled VLN)
#pragma unroll
    for (int l = 0; l < 10; ++l)
#pragma unroll
        for (int n = 0; n < 3; ++n) B2[l][n] = S2[l] * A2[l][n];

    // ---- layer 3 fused: z3 and A3 = W3 @ B2 (raw) ----
    float x3[5], S3[5], A3[5][3];
#pragma unroll
    for (int k = 0; k < 5; ++k) {
        float z = b3[k], a0 = 0.f, a1 = 0.f, a2 = 0.f;
#pragma unroll
        for (int l = 0; l < 10; ++l) {
            float w = W3[k][l];
            z  += w * x2[l];
            a0 += w * B2[l][0];
            a1 += w * B2[l][1];
            a2 += w * B2[l][2];
        }
        float th = fast_tanh(z);
        x3[k] = th; S3[k] = 1.0f - th * th;
        A3[k][0] = a0; A3[k][1] = a1; A3[k][2] = a2;
    }
    float B3[5][3];                        // B3 = S3 ⊙ A3
#pragma unroll
    for (int k = 0; k < 5; ++k)
#pragma unroll
        for (int n = 0; n < 3; ++n) B3[k][n] = S3[k] * A3[k][n];

    // ---- layer 4 fused: z4 and A4 = W4 @ B3 (raw) ----
    float x4[5], S4[5], A4[5][3];
#pragma unroll
    for (int j = 0; j < 5; ++j) {
        float z = b4[j], a0 = 0.f, a1 = 0.f, a2 = 0.f;
#pragma unroll
        for (int k = 0; k < 5; ++k) {
            float w = W4[j][k];
            z  += w * x3[k];
            a0 += w * B3[k][0];
            a1 += w * B3[k][1];
            a2 += w * B3[k][2];
        }
        float th = fast_tanh(z);
        x4[j] = th; S4[j] = 1.0f - th * th;
        A4[j][0] = a0; A4[j][1] = a1; A4[j][2] = a2;
    }

    // ---- gradient g[n] = Σ_j (W5[j] S4[j]) A4[j][n] ----
    float pS2[5];
#pragma unroll
    for (int j = 0; j < 5; ++j) pS2[j] = W5[j] * S4[j];
    float g0 = 0.f, g1 = 0.f, g2 = 0.f;
#pragma unroll
    for (int j = 0; j < 5; ++j) {
        g0 += pS2[j] * A4[j][0];
        g1 += pS2[j] * A4[j][1];
        g2 += pS2[j] * A4[j][2];
    }

    // ---- back-propagated row vectors ----
    float r2[5];                           // r2 = W4^T pS2
#pragma unroll
    for (int k = 0; k < 5; ++k) {
        float s = 0.f;
#pragma unroll
        for (int j = 0; j < 5; ++j) s += pS2[j] * W4[j][k];
        r2[k] = s;
    }
    float pS3[5];
#pragma unroll
    for (int k = 0; k < 5; ++k) pS3[k] = r2[k] * S3[k];
    float r3[10];                          // r3 = W3^T pS3
#pragma unroll
    for (int l = 0; l < 10; ++l) {
        float s = 0.f;
#pragma unroll
        for (int k = 0; k < 5; ++k) s += pS3[k] * W3[k][l];
        r3[l] = s;
    }
    float pS4[10];
#pragma unroll
    for (int l = 0; l < 10; ++l) pS4[l] = r3[l] * S2[l];
    float r4[10];                          // r4 = W2^T pS4
#pragma unroll
    for (int m = 0; m < 10; ++m) {
        float s = 0.f;
#pragma unroll
        for (int l = 0; l < 10; ++l) s += pS4[l] * W2[l][m];
        r4[m] = s;
    }

    // ---- Hessian entries as rank-1 quadratic sums ----
    float H00 = 0.f, H11 = 0.f, H01 = 0.f;
#pragma unroll
    for (int j = 0; j < 5; ++j) {          // T1
        float e = -2.0f * W5[j] * S4[j] * x4[j];
        float a0 = A4[j][0], a1 = A4[j][1];
        H00 += e * a0 * a0; H11 += e * a1 * a1; H01 += e * a0 * a1;
    }
#pragma unroll
    for (int k = 0; k < 5; ++k) {          // T2
        float e = -2.0f * r2[k] * S3[k] * x3[k];
        float a0 = A3[k][0], a1 = A3[k][1];
        H00 += e * a0 * a0; H11 += e * a1 * a1; H01 += e * a0 * a1;
    }
#pragma unroll
    for (int l = 0; l < 10; ++l) {         // T3
        float e = -2.0f * r3[l] * S2[l] * x2[l];
        float a0 = A2[l][0], a1 = A2[l][1];
        H00 += e * a0 * a0; H11 += e * a1 * a1; H01 += e * a0 * a1;
    }
#pragma unroll
    for (int m = 0; m < 10; ++m) {         // T4
        float c = -r4[m] * x1[m];
        float w0 = W1[m][0], w1 = W1[m][1];
        H00 += c * w0 * w0; H11 += c * w1 * w1; H01 += c * w0 * w1;
    }

    // ---- PDE residual (E_C=B_0=ASP=ALPH=1, N_EXP=21) ----
    const float Mx = g0, My = g1, Mt = g2;
    const float Jsq = Mx * Mx + My * My;
    const float q2 = Jsq * Jsq;
    const float q4 = q2 * q2;
    const float q8 = q4 * q4;
    const float J9 = q8 * Jsq;             // Jsq^9
    const float J10 = q8 * q2;             // Jsq^10
    const float term1 = J10 * (H00 + H11);
    const float inner = (2.0f * Mx * H00 + 2.0f * My * H01) * Mx +
                        (2.0f * Mx * H01 + 2.0f * My * H11) * My;
    const float res = Mt - (__sinf(tcoord) + term1 + 10.0f * J9 * inner);

    const long long gid = pbase + tid;
    if (gid < V) __builtin_nontemporal_store(res, &out[gid]);
}

extern "C" void kernel_launch(void* const* d_in, const int* in_sizes, int n_in,
                              void* d_out, int out_size, void* d_ws, size_t ws_size,
                              hipStream_t stream) {
    const float* X  = (const float*)d_in[0];
    const float* W1 = (const float*)d_in[1];
    const float* b1 = (const float*)d_in[2];
    const float* W2 = (const float*)d_in[3];
    const float* b2 = (const float*)d_in[4];
    const float* W3 = (const float*)d_in[5];
    const float* b3 = (const float*)d_in[6];
    const float* W4 = (const float*)d_in[7];
    const float* b4 = (const float*)d_in[8];
    const float* W5 = (const float*)d_in[9];
    // d_in[10] = b5: unused (constant offset vanishes under d/dx and d2/dx2)
    float* out = (float*)d_out;

    const int V    = in_sizes[0] / 3;
    const int grid = (V + BLOCK - 1) / BLOCK;
    pinn_pde_kernel<<<grid, BLOCK, 0, stream>>>(X, W1, b1, W2, b2, W3, b3, W4, b4,
                                                W5, out, V);
    (void)n_in; (void)out_size; (void)d_ws; (void)ws_size;
}